// GraphSAGELayer_22187801051307
// MI455X (gfx1250) — compile-verified
//
#include <hip/hip_runtime.h>
#include <math.h>

typedef __attribute__((ext_vector_type(2))) float v2f;
typedef __attribute__((ext_vector_type(4))) float v4f;
typedef __attribute__((ext_vector_type(8))) float v8f;

#define D 128
#define LNS 132              // padded LDS row stride (floats): rows land on distinct banks
#define WAVES_PER_BLOCK 4

__global__ __launch_bounds__(32 * WAVES_PER_BLOCK)
void sage_wmma_kernel(const float* __restrict__ h,
                      const int*   __restrict__ nbr_idx,
                      const float* __restrict__ nbr_w,
                      const float* __restrict__ w_self,
                      const float* __restrict__ w_nei,
                      const float* __restrict__ gamma,
                      const float* __restrict__ beta,
                      float* __restrict__ out,
                      int N, int K)
{
    __shared__ float lds[WAVES_PER_BLOCK * 16 * LNS];

    const int lane   = threadIdx.x & 31;
    const int waveId = threadIdx.x >> 5;
    const int tile   = blockIdx.x * WAVES_PER_BLOCK + waveId;
    if (tile * 16 >= N) return;          // wave-uniform: EXEC stays all-ones for WMMA

    const int m    = lane & 15;          // matrix row handled by this lane (frag reads)
    const int half = lane >> 4;          // 0: K sub-cols {0,1}; 1: {2,3}
    const int off  = half * 2;

    float* tileLds = lds + waveId * (16 * LNS);

    // ======== stage h self-rows, fully coalesced: one b128 per row ========
    for (int r = 0; r < 16; ++r) {
        int grow = tile * 16 + r;               // wave-uniform
        if (grow >= N) grow = N - 1;
        v4f v = *reinterpret_cast<const v4f*>(h + (size_t)grow * D + 4 * lane);
        *reinterpret_cast<v4f*>(tileLds + r * LNS + 4 * lane) = v;
    }

    // ---- A_self fragments (WMMA f32 16x16x4 A-layout) from LDS
    v2f aS[32];
#pragma unroll
    for (int c = 0; c < 32; ++c)
        aS[c] = *reinterpret_cast<const v2f*>(tileLds + m * LNS + 4 * c + off);

    // ======== fused gather + weighted aggregation, coalesced b128 rows ========
    // row index & weight are wave-uniform per (r,k) -> scalar loads
    for (int r = 0; r < 16; ++r) {
        int grow = tile * 16 + r;
        if (grow >= N) grow = N - 1;
        v4f acc = {};
        for (int k = 0; k < K; ++k) {
            const int   nrow = nbr_idx[(size_t)grow * K + k];   // uniform -> s_load
            const float w    = nbr_w [(size_t)grow * K + k];    // uniform -> s_load
            v4f v = *reinterpret_cast<const v4f*>(h + (size_t)nrow * D + 4 * lane);
            acc += v * w;
        }
        *reinterpret_cast<v4f*>(tileLds + r * LNS + 4 * lane) = acc;  // overwrite stage
    }

    // ---- A_agg fragments from LDS
    v2f aA[32];
#pragma unroll
    for (int c = 0; c < 32; ++c)
        aA[c] = *reinterpret_cast<const v2f*>(tileLds + m * LNS + 4 * c + off);

    // ======== dual GEMM: z = h @ Wself^T + agg @ Wnei^T (16x128 x 128x128) ========
    for (int jt = 0; jt < 8; ++jt) {
        const int j0 = jt * 16;
        v8f acc = {};
        // B[k][n] = W[j0+n][k0+k] -> per-lane float2 addressing on W rows
        const float* bsRow = w_self + (size_t)(j0 + m) * D + off;
        const float* bnRow = w_nei  + (size_t)(j0 + m) * D + off;
#pragma unroll
        for (int c = 0; c < 32; ++c) {
            v2f bS = *reinterpret_cast<const v2f*>(bsRow + 4 * c);
            acc = __builtin_amdgcn_wmma_f32_16x16x4_f32(false, aS[c], false, bS,
                                                        (short)0, acc, false, false);
            v2f bN = *reinterpret_cast<const v2f*>(bnRow + 4 * c);
            acc = __builtin_amdgcn_wmma_f32_16x16x4_f32(false, aA[c], false, bN,
                                                        (short)0, acc, false, false);
        }
        // C/D layout: VGPR r -> row (r + 8*half), col (j0 + m). Exact GELU, stash.
#pragma unroll
        for (int r = 0; r < 8; ++r) {
            const int rr = r + 8 * half;
            const float z = acc[r];
            const float g = 0.5f * z * (1.0f + erff(z * 0.70710678118654752f));
            tileLds[rr * LNS + j0 + m] = g;
        }
    }

    // ======== wave-cooperative LayerNorm over each of the 16 rows ========
    for (int r = 0; r < 16; ++r) {
        const int grow = tile * 16 + r;
        const float x0 = tileLds[r * LNS + lane];
        const float x1 = tileLds[r * LNS + lane + 32];
        const float x2 = tileLds[r * LNS + lane + 64];
        const float x3 = tileLds[r * LNS + lane + 96];
        float s = x0 + x1 + x2 + x3;
#pragma unroll
        for (int d = 16; d >= 1; d >>= 1) s += __shfl_xor(s, d, 32);
        const float mu = s * (1.0f / D);
        const float d0 = x0 - mu, d1 = x1 - mu, d2 = x2 - mu, d3 = x3 - mu;
        float s2 = d0 * d0 + d1 * d1 + d2 * d2 + d3 * d3;
#pragma unroll
        for (int d = 16; d >= 1; d >>= 1) s2 += __shfl_xor(s2, d, 32);
        const float inv = rsqrtf(s2 * (1.0f / D) + 1e-5f);
        if (grow < N) {
            float* o = out + (size_t)grow * D;
            o[lane]      = d0 * inv * gamma[lane]      + beta[lane];
            o[lane + 32] = d1 * inv * gamma[lane + 32] + beta[lane + 32];
            o[lane + 64] = d2 * inv * gamma[lane + 64] + beta[lane + 64];
            o[lane + 96] = d3 * inv * gamma[lane + 96] + beta[lane + 96];
        }
    }
}

extern "C" void kernel_launch(void* const* d_in, const int* in_sizes, int n_in,
                              void* d_out, int out_size, void* d_ws, size_t ws_size,
                              hipStream_t stream) {
    const float* h       = (const float*)d_in[0];
    const int*   nbr_idx = (const int*)  d_in[1];   // JAX canonicalizes int64 -> int32
    const float* nbr_w   = (const float*)d_in[2];
    const float* w_self  = (const float*)d_in[3];
    const float* w_nei   = (const float*)d_in[4];
    const float* gamma   = (const float*)d_in[5];
    const float* beta    = (const float*)d_in[6];
    float* out = (float*)d_out;

    const int N = in_sizes[0] / D;
    const int K = (N > 0) ? (in_sizes[1] / N) : 16;

    const int tiles  = (N + 15) / 16;
    const int blocks = (tiles + WAVES_PER_BLOCK - 1) / WAVES_PER_BLOCK;
    hipLaunchKernelGGL(sage_wmma_kernel, dim3(blocks), dim3(32 * WAVES_PER_BLOCK), 0, stream,
                       h, nbr_idx, nbr_w, w_self, w_nei, gamma, beta, out, N, K);
}